// TAG_NET_56891136803143
// MI455X (gfx1250) — compile-verified
//
#include <hip/hip_runtime.h>

#define N_NODES 100000
#define N_EDGES 1600000
#define FEAT 32
#define HID 64
#define CLS 10

typedef __attribute__((ext_vector_type(2))) float v2f;
typedef __attribute__((ext_vector_type(8))) float v8f;

// ---------------- utility kernels ----------------

__global__ void zero4_kernel(float4* __restrict__ p, long n4) {
  long i = (long)blockIdx.x * blockDim.x + threadIdx.x;
  if (i < n4) p[i] = make_float4(0.f, 0.f, 0.f, 0.f);
}

__global__ void degree_kernel(const int* __restrict__ col, float* __restrict__ deg) {
  int i = blockIdx.x * blockDim.x + threadIdx.x;
  if (i < N_EDGES) atomicAdd(&deg[col[i]], 1.0f);
}

__global__ void rsqrt_kernel(float* __restrict__ d) {
  int i = blockIdx.x * blockDim.x + threadIdx.x;
  if (i < N_NODES) {
    float v = d[i];
    d[i] = (v > 0.0f) ? rsqrtf(v) : 0.0f;
  }
}

__global__ void norm_kernel(const int* __restrict__ row, const int* __restrict__ col,
                            const float* __restrict__ dinv, float* __restrict__ nrm) {
  int i = blockIdx.x * blockDim.x + threadIdx.x;
  if (i < N_EDGES) nrm[i] = dinv[row[i]] * dinv[col[i]];
}

// One wave per edge; edge index made wave-uniform via readfirstlane so that
// row/col/norm become scalar loads. Feature gather is coalesced per lane.
// dst must be pre-zeroed:  dst[col[e], f] += norm[e] * src[row[e], f]
template <int WIDTH>  // 32 or 64
__global__ void scatter_edge_kernel(const float* __restrict__ src, float* __restrict__ dst,
                                    const int* __restrict__ row, const int* __restrict__ col,
                                    const float* __restrict__ nrm) {
  int wv = blockIdx.x * (blockDim.x >> 5) + (threadIdx.x >> 5);
  if (wv >= N_EDGES) return;  // wave-uniform exit
  const int lane = threadIdx.x & 31;
  const int e = __builtin_amdgcn_readfirstlane(wv);  // force wave-uniform -> scalar loads
  const int r = row[e];
  const int c = col[e];
  const float w = nrm[e];
  if (WIDTH == 32) {
    float v = w * src[(long)r * 32 + lane];
    atomicAdd(&dst[(long)c * 32 + lane], v);
  } else {
    const float2* s2 = (const float2*)(src + (long)r * WIDTH) + lane;  // b64 load
    float2 v = *s2;
    float* d = dst + (long)c * WIDTH + 2 * lane;
    atomicAdd(d + 0, w * v.x);
    atomicAdd(d + 1, w * v.y);
  }
}

// ---------------- WMMA fp32 GEMM ----------------
// C[M x NREAL] (+)= A[M x KDIM] * B[KDIM x NREAL]; M = N_NODES (multiple of 16).
// One wave computes a 16 x (NTILES*16) row-stripe: A fragment loaded once per
// K-step, reused across all NTILES WMMAs (V_WMMA_F32_16X16X4_F32, K-step 4).
// Optional fused bias+ReLU epilogue.
template <int KDIM, int NTILES, int NREAL, bool BIAS_RELU>
__global__ void gemm_wmma_f32(const float* __restrict__ A, const float* __restrict__ B,
                              float* __restrict__ C, const float* __restrict__ bias,
                              int accumulate) {
  const int lane = threadIdx.x & 31;
  int mt = blockIdx.x * (blockDim.x >> 5) + (threadIdx.x >> 5);
  if (mt >= N_NODES / 16) return;  // wave-uniform exit
  const int m0 = mt * 16;
  const int l16 = lane & 15;
  const int hi = lane >> 4;  // 0 or 1
  const int kh = hi * 2;     // K sub-offset within the 4-wide K step

  v8f acc[NTILES];
#pragma unroll
  for (int nt = 0; nt < NTILES; ++nt) {
    int n = nt * 16 + l16;
    bool ok = (n < NREAL);
    int nc = ok ? n : 0;
    if (accumulate) {
#pragma unroll
      for (int v = 0; v < 8; ++v) {
        int r = m0 + v + hi * 8;  // C layout: VGPR v holds row M = v + 8*(lane/16)
        float cv = C[(long)r * NREAL + nc];
        acc[nt][v] = ok ? cv : 0.0f;
      }
    } else {
#pragma unroll
      for (int v = 0; v < 8; ++v) acc[nt][v] = 0.0f;
    }
  }

  const float* arow = A + (long)(m0 + l16) * KDIM + kh;  // A: row m=lane%16, K pair by lane/16
#pragma unroll
  for (int k0 = 0; k0 < KDIM; k0 += 4) {
    v2f a;
    a.x = arow[k0];
    a.y = arow[k0 + 1];
#pragma unroll
    for (int nt = 0; nt < NTILES; ++nt) {
      int n = nt * 16 + l16;
      bool ok = (n < NREAL);
      int nc = ok ? n : 0;
      v2f b;
      float b0 = B[(k0 + kh) * NREAL + nc];
      float b1 = B[(k0 + kh + 1) * NREAL + nc];
      b.x = ok ? b0 : 0.0f;
      b.y = ok ? b1 : 0.0f;
      acc[nt] = __builtin_amdgcn_wmma_f32_16x16x4_f32(false, a, false, b, (short)0,
                                                      acc[nt], false, false);
    }
  }

#pragma unroll
  for (int nt = 0; nt < NTILES; ++nt) {
    int n = nt * 16 + l16;
    if (n < NREAL) {
      float bv = BIAS_RELU ? bias[n] : 0.0f;
#pragma unroll
      for (int v = 0; v < 8; ++v) {
        int r = m0 + v + hi * 8;
        float val = acc[nt][v];
        if (BIAS_RELU) {
          val += bv;
          val = (val > 0.0f) ? val : 0.0f;
        }
        C[(long)r * NREAL + n] = val;
      }
    }
  }
}

// ---------------- epilogue ----------------

__global__ void logsoftmax_kernel(float* __restrict__ out, const float* __restrict__ b2) {
  int i = blockIdx.x * blockDim.x + threadIdx.x;
  if (i >= N_NODES) return;
  float v[CLS];
  float mx = -1e30f;
#pragma unroll
  for (int c = 0; c < CLS; ++c) {
    v[c] = out[(long)i * CLS + c] + b2[c];
    mx = fmaxf(mx, v[c]);
  }
  float s = 0.0f;
#pragma unroll
  for (int c = 0; c < CLS; ++c) s += expf(v[c] - mx);
  float lse = mx + logf(s);
#pragma unroll
  for (int c = 0; c < CLS; ++c) out[(long)i * CLS + c] = v[c] - lse;
}

// ---------------- launcher ----------------

extern "C" void kernel_launch(void* const* d_in, const int* in_sizes, int n_in,
                              void* d_out, int out_size, void* d_ws, size_t ws_size,
                              hipStream_t stream) {
  const float* x  = (const float*)d_in[0];
  const int*   ei = (const int*)d_in[1];
  const float* W1 = (const float*)d_in[2];   // (4, 32, 64)
  const float* b1 = (const float*)d_in[3];   // (64,)
  const float* W2 = (const float*)d_in[4];   // (4, 64, 10)
  const float* b2 = (const float*)d_in[5];   // (10,)
  const int* row = ei;                        // edge_index[0]
  const int* col = ei + N_EDGES;              // edge_index[1]
  float* out = (float*)d_out;                 // (100000, 10)

  // workspace layout (floats): dinv | norm | bufA | bufB | h1   (~84 MB total)
  float* ws   = (float*)d_ws;
  float* dinv = ws;
  float* nrm  = dinv + N_NODES;
  float* bufA = nrm + N_EDGES;
  float* bufB = bufA + (long)N_NODES * HID;
  float* h1   = bufB + (long)N_NODES * HID;

  const int T = 256;
  auto nb = [](long n, int t) { return (int)((n + t - 1) / t); };

  const int scatter_blocks = nb((long)N_EDGES * 32, T);          // 1 wave / edge
  const int gemm_blocks = nb((long)(N_NODES / 16) * 32, T);      // 1 wave / 16-row stripe
  const long zf = (long)N_NODES * FEAT / 4;                      // float4 counts
  const long zh = (long)N_NODES * HID / 4;

  // ---- gcn_norm: deg -> dinv -> per-edge norm ----
  zero4_kernel<<<nb(N_NODES / 4, T), T, 0, stream>>>((float4*)dinv, N_NODES / 4);
  degree_kernel<<<nb(N_EDGES, T), T, 0, stream>>>(col, dinv);
  rsqrt_kernel<<<nb(N_NODES, T), T, 0, stream>>>(dinv);
  norm_kernel<<<nb(N_EDGES, T), T, 0, stream>>>(row, col, dinv, nrm);

  // ---- layer 1: 32 -> 64, K=3 hops; bias+ReLU fused into the last hop GEMM ----
  gemm_wmma_f32<FEAT, HID / 16, HID, false>
      <<<gemm_blocks, T, 0, stream>>>(x, W1, h1, nullptr, 0);

  zero4_kernel<<<nb(zf, T), T, 0, stream>>>((float4*)bufA, zf);
  scatter_edge_kernel<32><<<scatter_blocks, T, 0, stream>>>(x, bufA, row, col, nrm);
  gemm_wmma_f32<FEAT, HID / 16, HID, false>
      <<<gemm_blocks, T, 0, stream>>>(bufA, W1 + 1 * FEAT * HID, h1, nullptr, 1);

  zero4_kernel<<<nb(zf, T), T, 0, stream>>>((float4*)bufB, zf);
  scatter_edge_kernel<32><<<scatter_blocks, T, 0, stream>>>(bufA, bufB, row, col, nrm);
  gemm_wmma_f32<FEAT, HID / 16, HID, false>
      <<<gemm_blocks, T, 0, stream>>>(bufB, W1 + 2 * FEAT * HID, h1, nullptr, 1);

  zero4_kernel<<<nb(zf, T), T, 0, stream>>>((float4*)bufA, zf);
  scatter_edge_kernel<32><<<scatter_blocks, T, 0, stream>>>(bufB, bufA, row, col, nrm);
  gemm_wmma_f32<FEAT, HID / 16, HID, true>
      <<<gemm_blocks, T, 0, stream>>>(bufA, W1 + 3 * FEAT * HID, h1, b1, 1);

  // ---- layer 2: 64 -> 10, K=3 hops (N padded to one 16-wide tile, cols masked) ----
  gemm_wmma_f32<HID, 1, CLS, false>
      <<<gemm_blocks, T, 0, stream>>>(h1, W2, out, nullptr, 0);

  zero4_kernel<<<nb(zh, T), T, 0, stream>>>((float4*)bufA, zh);
  scatter_edge_kernel<64><<<scatter_blocks, T, 0, stream>>>(h1, bufA, row, col, nrm);
  gemm_wmma_f32<HID, 1, CLS, false>
      <<<gemm_blocks, T, 0, stream>>>(bufA, W2 + 1 * HID * CLS, out, nullptr, 1);

  zero4_kernel<<<nb(zh, T), T, 0, stream>>>((float4*)bufB, zh);
  scatter_edge_kernel<64><<<scatter_blocks, T, 0, stream>>>(bufA, bufB, row, col, nrm);
  gemm_wmma_f32<HID, 1, CLS, false>
      <<<gemm_blocks, T, 0, stream>>>(bufB, W2 + 2 * HID * CLS, out, nullptr, 1);

  zero4_kernel<<<nb(zh, T), T, 0, stream>>>((float4*)bufA, zh);
  scatter_edge_kernel<64><<<scatter_blocks, T, 0, stream>>>(bufB, bufA, row, col, nrm);
  gemm_wmma_f32<HID, 1, CLS, false>
      <<<gemm_blocks, T, 0, stream>>>(bufA, W2 + 3 * HID * CLS, out, nullptr, 1);

  logsoftmax_kernel<<<nb(N_NODES, T), T, 0, stream>>>(out, b2);
}